// SPN_Layer_60928406061384
// MI455X (gfx1250) — compile-verified
//
#include <hip/hip_runtime.h>
#include <hip/hip_bf16.h>

// ---------------------------------------------------------------------------
// SPN layer for MI455X (gfx1250, wave32).
//   agg = x + scatter_add(softmax(hop)[d-1] * x[col] -> row)
//   h   = relu(BN(agg @ W1 + b1));  h = relu(BN(h @ W2 + b2))
// Scatter phase is the HBM/L2 roofline limiter (~1.7 GB of traffic);
// GEMMs (2 x 100000x128x128 = 6.6 GFLOP, ~200 MB traffic) are memory-bound,
// so f32 WMMA (v_wmma_f32_16x16x4_f32) is the right precision: zero
// conversion cost, numerics match the reference.
// GEMM A-tiles are staged to LDS with the CDNA5 async global->LDS path
// (global_load_async_to_lds_b128 + s_wait_asynccnt), killing the 8x
// redundant A reads across the 8 waves of a block.
// ---------------------------------------------------------------------------

typedef __attribute__((ext_vector_type(2))) float v2f;
typedef __attribute__((ext_vector_type(8))) float v8f;

#define IC 128            // in/out channels (I == O == 128)
#define BN_EPS 1e-5f

// ---- softmax over hop coefficients (K <= 8), single lane ------------------
__global__ void hop_softmax_kernel(const float* __restrict__ hc,
                                   float* __restrict__ hopw, int K) {
    if (threadIdx.x == 0 && blockIdx.x == 0) {
        float mx = hc[0];
        for (int i = 1; i < K; ++i) mx = fmaxf(mx, hc[i]);
        float e[8];
        float s = 0.f;
        for (int i = 0; i < K; ++i) { e[i] = __expf(hc[i] - mx); s += e[i]; }
        float inv = 1.f / s;
        for (int i = 0; i < K; ++i) hopw[i] = e[i] * inv;
    }
}

// ---- agg init: agg = x (float4 copy) --------------------------------------
__global__ void copy4_kernel(const float4* __restrict__ src,
                             float4* __restrict__ dst, long n4) {
    long i = (long)blockIdx.x * blockDim.x + threadIdx.x;
    if (i < n4) dst[i] = src[i];
}

// ---- edge scatter: each thread = one edge x 4 features --------------------
__global__ void scatter_kernel(const int* __restrict__ ei,     // [2,E]
                               const int* __restrict__ ew,     // [E]
                               const float* __restrict__ x,    // [N,128]
                               const float* __restrict__ hopw, // [K]
                               float* __restrict__ agg,        // [N,128]
                               long E, int K) {
    long t = (long)blockIdx.x * blockDim.x + threadIdx.x;
    long e = t >> 5;                 // 32 threads per edge
    if (e >= E) return;
    int d = ew[e];
    if (d < 1 || d > K) return;
    float w = hopw[d - 1];
    int c   = (int)(t & 31) * 4;     // 4 contiguous features per thread
    int row = ei[e];
    int col = ei[E + e];
    float4 xv = *(const float4*)(x + (size_t)col * IC + c);
    float* dst = agg + (size_t)row * IC + c;
    atomicAdd(dst + 0, w * xv.x);
    atomicAdd(dst + 1, w * xv.y);
    atomicAdd(dst + 2, w * xv.z);
    atomicAdd(dst + 3, w * xv.w);
}

// ---- WMMA f32 GEMM:  Y[nrows,128] = A[nrows,128] @ W[128,128] + bias ------
// blockIdx.x = 16-row tile; 8 waves per block, wave w owns 16-col tile w.
// A tile (16x128 f32 = 8 KB) is staged in LDS via async global->LDS b128
// copies (2 issues per wave covering 4 KB each across the workgroup).
// A-frag (16x4 f32): lanes 0-15 -> K=k0,k0+1 ; lanes 16-31 -> K=k0+2,k0+3.
// B-frag (4x16 f32): VGPR0 = rows K=k0 / k0+2, VGPR1 = K=k0+1 / k0+3.
// C/D (16x16 f32):  VGPR r holds M=r (lanes 0-15) and M=r+8 (lanes 16-31).
__global__ __launch_bounds__(256)
void gemm_bias_kernel(const float* __restrict__ A,
                      const float* __restrict__ W,
                      const float* __restrict__ bias,
                      float* __restrict__ Y, int nrows) {
    __shared__ float As[16 * IC];           // 8 KB A tile

    const int tid  = threadIdx.x;
    const int wave = tid >> 5;              // 0..7  -> column tile
    const int lane = tid & 31;
    const int row0 = blockIdx.x * 16;

    // ---- async-stage the A tile: 512 chunks of 16 B -----------------------
    const unsigned lds_base = (unsigned)(uintptr_t)As;
    #pragma unroll
    for (int pass = 0; pass < 2; ++pass) {
        int idx = pass * 256 + tid;         // chunk id in [0,512)
        int rin = idx >> 5;                 // row within tile (32 chunks/row)
        int srow = row0 + rin;
        if (srow >= nrows) srow = nrows - 1;          // clamp ragged tile
        const float* gp = A + (size_t)srow * IC + (idx & 31) * 4;
        unsigned lds_off = lds_base + (unsigned)idx * 16u;
        asm volatile("global_load_async_to_lds_b128 %0, %1, off"
                     :: "v"(lds_off), "v"(gp) : "memory");
    }
    asm volatile("s_wait_asynccnt 0x0" ::: "memory");
    __syncthreads();

    // ---- WMMA main loop ---------------------------------------------------
    const int m   = lane & 15;
    const int hk  = (lane >> 4) * 2;        // 0 (lanes 0-15) or 2 (16-31)
    const int col = wave * 16 + (lane & 15);
    const float* __restrict__ ap = &As[m * IC];

    v8f acc = {};
    #pragma unroll 4
    for (int k0 = 0; k0 < IC; k0 += 4) {
        const int kk = k0 + hk;
        v2f a, b;
        a.x = ap[kk];                       // ds_load_b64 (8 B aligned)
        a.y = ap[kk + 1];
        b.x = W[(size_t)(kk)     * IC + col];
        b.y = W[(size_t)(kk + 1) * IC + col];
        acc = __builtin_amdgcn_wmma_f32_16x16x4_f32(
                  /*neg_a=*/false, a, /*neg_b=*/false, b,
                  /*c_mod=*/(short)0, acc, /*reuse_a=*/false, /*reuse_b=*/false);
    }

    // ---- epilogue: bias + store (fast unguarded path for full tiles) ------
    const float bv = bias[col];
    const int rbase = row0 + ((lane >> 4) << 3);     // +8 rows, upper half-wave
    float* __restrict__ yp = Y + (size_t)rbase * IC + col;
    if (row0 + 16 <= nrows) {
        #pragma unroll
        for (int r = 0; r < 8; ++r) yp[(size_t)r * IC] = acc[r] + bv;
    } else {
        #pragma unroll
        for (int r = 0; r < 8; ++r)
            if (rbase + r < nrows) yp[(size_t)r * IC] = acc[r] + bv;
    }
}

// ---- zero the 256-float stats buffer --------------------------------------
__global__ void zero_stats_kernel(float* __restrict__ stats) {
    stats[threadIdx.x] = 0.f;
}

// ---- per-column sum / sum-of-squares (register partials + atomics) --------
__global__ void col_stats_kernel(const float* __restrict__ X,
                                 float* __restrict__ stats,
                                 int nrows, int rows_per_block) {
    const int col = threadIdx.x;            // blockDim.x == 128
    const int r0  = blockIdx.x * rows_per_block;
    int r1 = r0 + rows_per_block;
    if (r1 > nrows) r1 = nrows;
    float s = 0.f, sq = 0.f;
    for (int r = r0; r < r1; ++r) {
        float v = X[(size_t)r * IC + col];
        s  += v;
        sq += v * v;
    }
    atomicAdd(&stats[col], s);
    atomicAdd(&stats[IC + col], sq);
}

// ---- fold stats into per-column scale/shift -------------------------------
__global__ void bn_finalize_kernel(const float* __restrict__ stats,
                                   const float* __restrict__ gamma,
                                   const float* __restrict__ beta,
                                   float* __restrict__ ss, float inv_n) {
    const int c = threadIdx.x;              // blockDim.x == 128
    float mean = stats[c] * inv_n;
    float var  = stats[IC + c] * inv_n - mean * mean;
    float sc   = gamma[c] * rsqrtf(var + BN_EPS);
    ss[c]      = sc;
    ss[IC + c] = beta[c] - mean * sc;
}

// ---- y = relu(x * scale[col] + shift[col]), float4-vectorized -------------
__global__ void bn_relu4_kernel(const float4* __restrict__ X,
                                const float* __restrict__ ss,
                                float4* __restrict__ Y, long n4) {
    long i = (long)blockIdx.x * blockDim.x + threadIdx.x;
    if (i >= n4) return;
    int c = (int)((i * 4) & (IC - 1));
    float4 v = X[i];
    v.x = fmaxf(fmaf(v.x, ss[c + 0], ss[IC + c + 0]), 0.f);
    v.y = fmaxf(fmaf(v.y, ss[c + 1], ss[IC + c + 1]), 0.f);
    v.z = fmaxf(fmaf(v.z, ss[c + 2], ss[IC + c + 2]), 0.f);
    v.w = fmaxf(fmaf(v.w, ss[c + 3], ss[IC + c + 3]), 0.f);
    Y[i] = v;
}

extern "C" void kernel_launch(void* const* d_in, const int* in_sizes, int n_in,
                              void* d_out, int out_size, void* d_ws, size_t ws_size,
                              hipStream_t stream) {
    // setup_inputs() order
    const float* x      = (const float*)d_in[0];   // [N,128]
    const int*   ei     = (const int*)  d_in[1];   // [2,E]
    const int*   ew     = (const int*)  d_in[2];   // [E]
    const float* hc     = (const float*)d_in[3];   // [K]
    const float* W1     = (const float*)d_in[4];
    const float* b1     = (const float*)d_in[5];
    const float* g1     = (const float*)d_in[6];
    const float* be1    = (const float*)d_in[7];
    const float* W2     = (const float*)d_in[8];
    const float* b2     = (const float*)d_in[9];
    const float* g2     = (const float*)d_in[10];
    const float* be2    = (const float*)d_in[11];
    float*       out    = (float*)d_out;

    const int  N = in_sizes[0] / IC;
    const long E = (long)in_sizes[1] / 2;
    const int  K = in_sizes[3];

    // workspace layout: [hopw 8f][stats 256f][ss 256f][pad..4KB][agg N*128 f]
    char*  ws    = (char*)d_ws;
    float* hopw  = (float*)(ws);
    float* stats = (float*)(ws + 32);
    float* ss    = (float*)(ws + 32 + 1024);
    float* agg   = (float*)(ws + 4096);            // 51.2 MB for N=100000

    const long total = (long)N * IC;
    const long n4    = total / 4;
    const float inv_n = 1.f / (float)N;

    // 1) hop softmax
    hop_softmax_kernel<<<1, 32, 0, stream>>>(hc, hopw, K);

    // 2) agg = x
    copy4_kernel<<<(int)((n4 + 255) / 256), 256, 0, stream>>>(
        (const float4*)x, (float4*)agg, n4);

    // 3) edge scatter (32 threads/edge, 4 features each)
    {
        long nthr = E * 32;
        scatter_kernel<<<(int)((nthr + 255) / 256), 256, 0, stream>>>(
            ei, ew, x, hopw, agg, E, K);
    }

    const int row_tiles = (N + 15) / 16;

    // 4) y1 = agg @ W1 + b1   -> d_out (intermediate)
    gemm_bias_kernel<<<row_tiles, 256, 0, stream>>>(agg, W1, b1, out, N);

    // 5-7) BN1 stats + finalize
    zero_stats_kernel<<<1, 256, 0, stream>>>(stats);
    {
        const int rpb = 64;
        col_stats_kernel<<<(N + rpb - 1) / rpb, IC, 0, stream>>>(out, stats, N, rpb);
    }
    bn_finalize_kernel<<<1, IC, 0, stream>>>(stats, g1, be1, ss, inv_n);

    // 8) h1 = relu(BN(y1)) -> agg (reuse)
    bn_relu4_kernel<<<(int)((n4 + 255) / 256), 256, 0, stream>>>(
        (const float4*)out, ss, (float4*)agg, n4);

    // 9) y2 = h1 @ W2 + b2 -> d_out
    gemm_bias_kernel<<<row_tiles, 256, 0, stream>>>(agg, W2, b2, out, N);

    // 10-12) BN2 stats + finalize
    zero_stats_kernel<<<1, 256, 0, stream>>>(stats);
    {
        const int rpb = 64;
        col_stats_kernel<<<(N + rpb - 1) / rpb, IC, 0, stream>>>(out, stats, N, rpb);
    }
    bn_finalize_kernel<<<1, IC, 0, stream>>>(stats, g2, be2, ss, inv_n);

    // 13) out = relu(BN(y2)) in place
    bn_relu4_kernel<<<(int)((n4 + 255) / 256), 256, 0, stream>>>(
        (const float4*)out, ss, (float4*)out, n4);
}